// SummedBox2d_19997367730745
// MI455X (gfx1250) — compile-verified
//
#include <hip/hip_runtime.h>

#ifndef __has_builtin
#define __has_builtin(x) 0
#endif

#define BOX_R 7
#define BOX_W 15                        // 2*R+1
#define TILE 64                         // output tile edge
#define HALO (TILE + 2 * BOX_R)         // 78 rows loaded
#define HALOW 80                        // 80 cols loaded: [gx0-8, gx0+72)
#define VECW (HALOW / 4)                // 20 vec4s per halo row
#define SPITCH 84                       // s_in pitch (floats): 336B, 16B-aligned rows
#define HPITCH 65                       // s_h pitch: conflict-free stage2 W / stage3 R
#define IMG 512

typedef int v4i_vec __attribute__((vector_size(16)));

// ---- CDNA5 async global->LDS copies (ASYNCcnt path) --------------------
__device__ __forceinline__ void async_copy_b128(const float* g, float* l) {
#if __has_builtin(__builtin_amdgcn_global_load_async_to_lds_b128)
  __builtin_amdgcn_global_load_async_to_lds_b128(
      (__attribute__((address_space(1))) v4i_vec*)g,
      (__attribute__((address_space(3))) v4i_vec*)l, /*offset=*/0, /*cpol=*/0);
#elif __has_builtin(__builtin_amdgcn_global_load_async_to_lds_b32)
  // INST_OFFSET is added to BOTH the global and LDS address (ISA 08 §4.4),
  // so immediate offsets implement the 16-byte copy exactly.
  __builtin_amdgcn_global_load_async_to_lds_b32(
      (__attribute__((address_space(1))) int*)g,
      (__attribute__((address_space(3))) int*)l, 0, 0);
  __builtin_amdgcn_global_load_async_to_lds_b32(
      (__attribute__((address_space(1))) int*)g,
      (__attribute__((address_space(3))) int*)l, 4, 0);
  __builtin_amdgcn_global_load_async_to_lds_b32(
      (__attribute__((address_space(1))) int*)g,
      (__attribute__((address_space(3))) int*)l, 8, 0);
  __builtin_amdgcn_global_load_async_to_lds_b32(
      (__attribute__((address_space(1))) int*)g,
      (__attribute__((address_space(3))) int*)l, 12, 0);
#else
  unsigned lds_off =
      (unsigned)(__SIZE_TYPE__)(__attribute__((address_space(3))) void*)l;
  asm volatile("global_load_async_to_lds_b128 %0, %1, off"
               :
               : "v"(lds_off), "v"(g)
               : "memory");
#endif
}

__device__ __forceinline__ void wait_asynccnt0() {
#if __has_builtin(__builtin_amdgcn_s_wait_asynccnt)
  __builtin_amdgcn_s_wait_asynccnt(0);
#else
  asm volatile("s_wait_asynccnt 0" ::: "memory");
#endif
}

// ---- fused separable 15x15 box sum -------------------------------------
__global__ __launch_bounds__(256) void box15_fused(
    const float* __restrict__ in, float* __restrict__ out) {
  // s_in col c holds global col (gx0 - 8 + c); output x uses cols x+1 .. x+15.
  __shared__ __align__(16) float s_in[HALO][SPITCH];
  __shared__ float s_h[HALO][HPITCH];    // horizontal 15-tap sums

  const int tid = threadIdx.x;
  const int gx0 = blockIdx.x * TILE;
  const int gy0 = blockIdx.y * TILE;
  const float* gplane = in + (size_t)blockIdx.z * (IMG * IMG);
  float* oplane = out + (size_t)blockIdx.z * (IMG * IMG);

  // Stage 1: 78x80 halo tile -> LDS in 16B async DMA chunks; OOB vec4 -> 0.
  // gx is always a multiple of 4, so a vec4 is entirely in- or out-of-range.
  for (int e = tid; e < HALO * VECW; e += 256) {
    const int ly = e / VECW;
    const int lvx = e - ly * VECW;
    const int gy = gy0 + ly - BOX_R;
    const int gx = gx0 - 8 + lvx * 4;
    float* lp = &s_in[ly][lvx * 4];
    if ((unsigned)gy < IMG && (unsigned)gx < IMG) {
      async_copy_b128(gplane + (size_t)gy * IMG + gx, lp);
    } else {
      *(float4*)lp = make_float4(0.f, 0.f, 0.f, 0.f);
    }
  }
  wait_asynccnt0();   // each wave drains its own ASYNCcnt
  __syncthreads();    // then make all LDS writes workgroup-visible

  // Stage 2: horizontal running 15-tap sums.
  // Work item = (row y in [0,78), 16-wide x-chunk); 78*4 = 312 items.
  for (int w = tid; w < HALO * (TILE / 16); w += 256) {
    const int y = w >> 2;
    const int x0 = (w & 3) * 16;
    float s = 0.0f;
#pragma unroll
    for (int d = 1; d < BOX_W; ++d) s += s_in[y][x0 + d];
#pragma unroll
    for (int i = 0; i < 16; ++i) {
      s += s_in[y][x0 + i + BOX_W];
      s_h[y][x0 + i] = s;
      s -= s_in[y][x0 + i + 1];
    }
  }
  __syncthreads();

  // Stage 3: vertical running 15-tap sums + coalesced stores.
  // 256 threads = 64 columns x 4 row-chunks of 16.
  const int x = tid & (TILE - 1);
  const int y0 = (tid >> 6) * 16;
  float s = 0.0f;
#pragma unroll
  for (int d = 0; d < BOX_W - 1; ++d) s += s_h[y0 + d][x];
  float* orow = oplane + (size_t)(gy0 + y0) * IMG + gx0 + x;
#pragma unroll
  for (int i = 0; i < 16; ++i) {
    s += s_h[y0 + i + BOX_W - 1][x];
    orow[(size_t)i * IMG] = s;
    s -= s_h[y0 + i][x];
  }
}

extern "C" void kernel_launch(void* const* d_in, const int* in_sizes, int n_in,
                              void* d_out, int out_size, void* d_ws,
                              size_t ws_size, hipStream_t stream) {
  (void)n_in; (void)out_size; (void)d_ws; (void)ws_size;
  const float* x = (const float*)d_in[0];
  float* out = (float*)d_out;
  const int planes = in_sizes[0] / (IMG * IMG);  // 8*1*16 = 128
  dim3 grid(IMG / TILE, IMG / TILE, planes);     // (8, 8, 128)
  box15_fused<<<grid, dim3(256, 1, 1), 0, stream>>>(x, out);
}